// CrossAttention_3925600108624
// MI455X (gfx1250) — compile-verified
//
#include <hip/hip_runtime.h>
#include <hip/hip_bf16.h>

// CrossAttention: out[b,n,m] = softmax_m( -||Q[b,n]-K[b,m]|| / 8 )
// B=4, N=M=4096, D=64, fp32 in/out.
//
// MI455X (gfx1250, wave32) plan:
//  - d2 = q^2 + k^2 - 2 q.k via chained V_WMMA_F32_16X16X4_F32
//    (A pre-scaled by -2, accumulator seeded with q^2+k^2).
//  - 2-way column split per row-tile (2048 waves -> ~2/SIMD) with the
//    per-row (max,sum) merged across halves through LDS between passes.
//  - Pass 1: online per-lane (max,sum) in exp2 domain; pass 2 recomputes the
//    tile and writes normalized probs once with nontemporal stores (256 MB =
//    the HBM floor; K re-reads stay in the 192 MB L2, kept hot by NT stores).
//  - K tile (16 cols x 64) staged to LDS pre-swizzled into B-fragment layout.

typedef __attribute__((ext_vector_type(2))) float v2f;
typedef __attribute__((ext_vector_type(4))) float v4f;
typedef __attribute__((ext_vector_type(8))) float v8f;

#define BATCH 4
#define NDIM  4096
#define DDIM  64
#define HALF_COLS (NDIM / 2)                 // 2048 columns per wave
#define TILES_PER_HALF (HALF_COLS / 16)      // 128
#define ROWS_PER_BLOCK 64                    // 4 row-tiles x 16
#define BLOCKS_PER_BATCH (NDIM / ROWS_PER_BLOCK)   // 64 -> grid 256
#define CL2E 0.180336880111113703f           // log2(e) / 8

__global__ __launch_bounds__(256)
void xattn_cdist_softmax(const float* __restrict__ Q,
                         const float* __restrict__ K,
                         float* __restrict__ out)
{
    // per column-half: B fragments (16 k-steps x 32 lanes) + ||k||^2
    __shared__ v2f   ldsB[2][16 * 32];          // 8 KB
    __shared__ float ldsK2[2][16];
    __shared__ float ldsM[2][ROWS_PER_BLOCK];   // per-half row max  (log2 dom)
    __shared__ float ldsS[2][ROWS_PER_BLOCK];   // per-half row sum

    const int t    = threadIdx.x;
    const int lane = t & 31;
    const int wave = t >> 5;
    const int g    = lane >> 4;                 // half-wave group (0/1)
    const int lh   = lane & 15;
    const int rt   = wave >> 1;                 // row-tile in block (0..3)
    const int h    = wave & 1;                  // column half (0/1)

    const int batch   = blockIdx.x / BLOCKS_PER_BATCH;
    const int rowBase = (blockIdx.x % BLOCKS_PER_BATCH) * ROWS_PER_BLOCK + rt * 16;
    const int colBase = h * HALF_COLS;

    const float* Qb = Q   + (size_t)batch * NDIM * DDIM;
    const float* Kb = K   + (size_t)batch * NDIM * DDIM;
    float*       Ob = out + (size_t)batch * NDIM * NDIM;

    // ---- A fragments: 16 rows x K=64, scaled by -2 (ISA 7.12.2 f32 A 16x4:
    // lanes 0-15 carry kdims {4k,4k+1}, lanes 16-31 carry {4k+2,4k+3}).
    const int arow = rowBase + lh;
    const int koff = 2 * g;
    v2f a[16];
    float q2p = 0.f;
#pragma unroll
    for (int k = 0; k < 16; ++k) {
        v2f qv = *(const v2f*)(Qb + (size_t)arow * DDIM + 4 * k + koff);
        q2p += qv.x * qv.x + qv.y * qv.y;
        a[k] = qv * (-2.0f);
    }
    float q2full = q2p + __shfl_xor(q2p, 16, 32);
    float q2v[8];                                // q^2 of rows rowBase+8g+v
#pragma unroll
    for (int v = 0; v < 8; ++v)
        q2v[v] = __shfl(q2full, 8 * g + v, 32);

    // staging role within this column-half group (4 waves = 128 threads):
    // thread sIdx stages col scol, fragment pair {sj, sj+1}
    const int sIdx = rt * 32 + lane;             // 0..127
    const int scol = sIdx >> 3;                  // 0..15
    const int sj   = (sIdx & 7) * 2;             // 0,2,..,14

    float lmax[8], lsum[8];
#pragma unroll
    for (int v = 0; v < 8; ++v) { lmax[v] = -1e30f; lsum[v] = 0.f; }

    for (int pass = 0; pass < 2; ++pass) {
        for (int tile = 0; tile < TILES_PER_HALF; ++tile) {
            __syncthreads();                     // previous tile fully consumed
            {
                const int col = colBase + tile * 16 + scol;
                const float* kp = Kb + (size_t)col * DDIM + 4 * sj;
                v4f k0 = *(const v4f*)(kp);
                v4f k1 = *(const v4f*)(kp + 4);
                // B 4x16 layout: group0 lanes take kdims {4k,4k+2} = (x,z),
                // group1 lanes take {4k+1,4k+3} = (y,w)
                v2f b; 
                b.x = k0.x; b.y = k0.z; ldsB[h][sj * 32 + scol]            = b;
                b.x = k0.y; b.y = k0.w; ldsB[h][sj * 32 + 16 + scol]       = b;
                b.x = k1.x; b.y = k1.z; ldsB[h][(sj + 1) * 32 + scol]      = b;
                b.x = k1.y; b.y = k1.w; ldsB[h][(sj + 1) * 32 + 16 + scol] = b;
                // ||k||^2: 8 staging threads per column, contiguous lanes
                float sq = k0.x*k0.x + k0.y*k0.y + k0.z*k0.z + k0.w*k0.w
                         + k1.x*k1.x + k1.y*k1.y + k1.z*k1.z + k1.w*k1.w;
                sq += __shfl_xor(sq, 1, 32);
                sq += __shfl_xor(sq, 2, 32);
                sq += __shfl_xor(sq, 4, 32);
                if ((sIdx & 7) == 0) ldsK2[h][scol] = sq;
                __builtin_prefetch(kp + 16 * DDIM, 0, 2);  // next K tile -> L2
            }
            __syncthreads();

            const float k2 = ldsK2[h][lh];
            v8f c;
#pragma unroll
            for (int v = 0; v < 8; ++v) c[v] = q2v[v] + k2;  // seed q^2+k^2

#pragma unroll
            for (int k = 0; k < 16; ++k) {
                v2f bb = ldsB[h][k * 32 + lane];
                c = __builtin_amdgcn_wmma_f32_16x16x4_f32(
                        false, a[k], false, bb, (short)0, c, false, false);
            }

            if (pass == 0) {
#pragma unroll
                for (int v = 0; v < 8; ++v) {
                    // score in log2 domain: -sqrt(d2) * log2(e)/8
                    float x  = -__builtin_amdgcn_sqrtf(fmaxf(c[v], 0.f)) * CL2E;
                    float nm = fmaxf(lmax[v], x);
                    lsum[v]  = lsum[v] * __builtin_amdgcn_exp2f(lmax[v] - nm)
                             + __builtin_amdgcn_exp2f(x - nm);
                    lmax[v]  = nm;
                }
            } else {
                const int col = colBase + tile * 16 + lh;
#pragma unroll
                for (int v = 0; v < 8; ++v) {
                    float x = -__builtin_amdgcn_sqrtf(fmaxf(c[v], 0.f)) * CL2E;
                    float p = __builtin_amdgcn_exp2f(x - lmax[v]) * lsum[v];
                    // dead data: NT store keeps K hot in L2
                    __builtin_nontemporal_store(
                        p, Ob + (size_t)(rowBase + 8 * g + v) * NDIM + col);
                }
            }
        }

        if (pass == 0) {
            // 1) merge (max,sum) across the 16 lanes of each half-wave group
#pragma unroll
            for (int off = 1; off < 16; off <<= 1) {
#pragma unroll
                for (int v = 0; v < 8; ++v) {
                    float om = __shfl_xor(lmax[v], off, 32);
                    float os = __shfl_xor(lsum[v], off, 32);
                    float nm = fmaxf(lmax[v], om);
                    lsum[v]  = lsum[v] * __builtin_amdgcn_exp2f(lmax[v] - nm)
                             + os      * __builtin_amdgcn_exp2f(om      - nm);
                    lmax[v]  = nm;
                }
            }
            // 2) publish this half's per-row (m,s); merge the two halves
            if (lh == 0) {
#pragma unroll
                for (int v = 0; v < 8; ++v) {
                    ldsM[h][rt * 16 + 8 * g + v] = lmax[v];
                    ldsS[h][rt * 16 + 8 * g + v] = lsum[v];
                }
            }
            __syncthreads();
#pragma unroll
            for (int v = 0; v < 8; ++v) {
                const int r = rt * 16 + 8 * g + v;
                float m0 = ldsM[0][r], s0 = ldsS[0][r];
                float m1 = ldsM[1][r], s1 = ldsS[1][r];
                float nm = fmaxf(m0, m1);
                float s  = s0 * __builtin_amdgcn_exp2f(m0 - nm)
                         + s1 * __builtin_amdgcn_exp2f(m1 - nm);
                lmax[v] = nm;
                lsum[v] = 1.0f / s;              // pass 2 uses reciprocal
            }
        }
    }
}

extern "C" void kernel_launch(void* const* d_in, const int* in_sizes, int n_in,
                              void* d_out, int out_size, void* d_ws, size_t ws_size,
                              hipStream_t stream) {
    (void)in_sizes; (void)n_in; (void)out_size; (void)d_ws; (void)ws_size;
    const float* Q = (const float*)d_in[0];
    const float* K = (const float*)d_in[1];
    float*       O = (float*)d_out;
    dim3 grid(BATCH * BLOCKS_PER_BATCH);   // 256 blocks
    dim3 block(256);                        // 8 wave32 per block
    xattn_cdist_softmax<<<grid, block, 0, stream>>>(Q, K, O);
}